// SiameseLoss_41497974014033
// MI455X (gfx1250) — compile-verified
//
#include <hip/hip_runtime.h>

typedef __attribute__((ext_vector_type(2))) float v2f;
typedef __attribute__((ext_vector_type(4))) float v4f;
typedef __attribute__((ext_vector_type(8))) float v8f;

#define DIMS 1024
#define MARGIN_F 1.0f

// --- Kernel 0: zero the scalar accumulator (harness poisons d_out) ---------
__global__ void SiameseZeroKernel(float* out) {
    if (threadIdx.x == 0 && blockIdx.x == 0) out[0] = 0.0f;
}

// --- Kernel 1: per-row inverse L2 norms (one wave32 per row) ---------------
__global__ void SiameseInvNormKernel(const float* __restrict__ emb,
                                     float* __restrict__ invn, int nrows) {
    const int lane = threadIdx.x & 31;
    const int row  = (int)((blockIdx.x * blockDim.x + threadIdx.x) >> 5);
    if (row >= nrows) return;
    const float* r = emb + (size_t)row * DIMS;
    float s = 0.0f;
#pragma unroll
    for (int k = 0; k < DIMS; k += 128) {              // 32 lanes * float4 = 128/iter
        v4f v = *(const v4f*)(r + k + lane * 4);
        s = fmaf(v.x, v.x, s); s = fmaf(v.y, v.y, s);
        s = fmaf(v.z, v.z, s); s = fmaf(v.w, v.w, s);
    }
#pragma unroll
    for (int off = 16; off > 0; off >>= 1) s += __shfl_xor(s, off, 32);
    if (lane == 0) {
        float g = rsqrtf(s);
        g = g * (1.5f - 0.5f * s * g * g);             // one Newton step
        invn[row] = g;
    }
}

// --- Kernel 2: 16 pairs per wave via V_WMMA_F32_16X16X4_F32 ----------------
// A fragment (16x4 f32): lane L holds {K=koff, K=koff+1} of row M=L&15, koff=(L>=16)?2:0
// B fragment (4x16 f32): identical per-lane load pattern (mirrored striping).
// Diagonal of C: lanes 0-7 -> c[lane], lanes 24-31 -> c[lane-24]; pair id = lane&15.
__global__ void SiamesePairWmmaKernel(const float* __restrict__ emb,
                                      const int* __restrict__ i1,
                                      const int* __restrict__ i2,
                                      const int* __restrict__ tgt,
                                      const float* __restrict__ invn,
                                      float* __restrict__ out,
                                      int npairs, float inv_p) {
    __shared__ float wsum[8];
    const int lane   = threadIdx.x & 31;
    const int waveIB = threadIdx.x >> 5;
    const int wave   = blockIdx.x * 8 + waveIB;
    const int p0     = wave * 16;

    float loss = 0.0f;
    if (p0 < npairs) {                                  // wave-uniform guard
        const int m    = lane & 15;
        const int rowA = i1[p0 + m];
        const int rowB = i2[p0 + m];
        const int koff = (lane >> 4) << 1;              // 0 or 2
        const float* aptr = emb + (size_t)rowA * DIMS + koff;
        const float* bptr = emb + (size_t)rowB * DIMS + koff;

        v8f c = {};
#pragma unroll 8
        for (int k = 0; k < DIMS; k += 4) {
            v2f a = *(const v2f*)(aptr + k);
            v2f b = *(const v2f*)(bptr + k);
            c = __builtin_amdgcn_wmma_f32_16x16x4_f32(
                    /*neg_a=*/false, a, /*neg_b=*/false, b,
                    /*c_mod=*/(short)0, c, /*reuse_a=*/false, /*reuse_b=*/false);
        }

        const bool valid = (lane < 8) || (lane >= 24);  // lanes holding diag(C)
        if (valid) {
            float dot   = c[lane & 7];
            float dist  = 1.0f - dot * invn[rowA] * invn[rowB];
            float t     = (float)tgt[p0 + m];
            float hinge = fmaxf(MARGIN_F - dist, 0.0f);
            loss = fmaf(t, dist - hinge, hinge);        // t*dist + (1-t)*hinge
        }
    }
#pragma unroll
    for (int off = 16; off > 0; off >>= 1) loss += __shfl_xor(loss, off, 32);
    if (lane == 0) wsum[waveIB] = loss;
    __syncthreads();
    if (threadIdx.x == 0) {
        float bs = 0.0f;
#pragma unroll
        for (int w = 0; w < 8; ++w) bs += wsum[w];
        atomicAdd(out, bs * inv_p);
    }
}

extern "C" void kernel_launch(void* const* d_in, const int* in_sizes, int n_in,
                              void* d_out, int out_size, void* d_ws, size_t ws_size,
                              hipStream_t stream) {
    const float* emb = (const float*)d_in[0];
    const int*   i1  = (const int*)d_in[1];
    const int*   i2  = (const int*)d_in[2];
    const int*   tg  = (const int*)d_in[3];
    float*       out = (float*)d_out;
    float*       invn = (float*)d_ws;                   // nrows * 4 bytes (32 KB)

    const int nrows = in_sizes[0] / DIMS;               // 8192
    const int P     = in_sizes[1];                      // 65536

    SiameseZeroKernel<<<1, 64, 0, stream>>>(out);

    const int nblk1 = (nrows * 32 + 255) / 256;         // one wave per row
    SiameseInvNormKernel<<<nblk1, 256, 0, stream>>>(emb, invn, nrows);

    const int waves2 = (P + 15) / 16;                   // 16 pairs per wave
    const int nblk2  = (waves2 + 7) / 8;                // 8 waves per block
    SiamesePairWmmaKernel<<<nblk2, 256, 0, stream>>>(emb, i1, i2, tg, invn, out,
                                                     P, 1.0f / (float)P);
}